// Siamese_87428354278299
// MI455X (gfx1250) — compile-verified
//
#include <hip/hip_runtime.h>
#include <hip/hip_bf16.h>

// Problem sizes (fixed by the reference)
#define NB 4      // batch
#define NM 1128   // samples / distance features
#define NL 512    // sequence length (softmax axis)
#define NFT 13    // MFCC dim
#define XSTR 18   // LDS row stride in dwords: even (8B-aligned b64) and 18*lane%64 is a permutation -> no bank conflicts

typedef __attribute__((ext_vector_type(2))) float v2f;
typedef __attribute__((ext_vector_type(8))) float v8f;

__device__ __forceinline__ v8f wmma_f32_16x16x4(v2f a, v2f b, v8f c) {
  // 8-arg pattern: (neg_a, A, neg_b, B, c_mod, C, reuse_a, reuse_b)
  return __builtin_amdgcn_wmma_f32_16x16x4_f32(false, a, false, b, (short)0, c,
                                               false, false);
}

#if __has_builtin(__builtin_amdgcn_tanhf)
__device__ __forceinline__ float fast_tanh(float x) {
  return __builtin_amdgcn_tanhf(x);   // native V_TANH_F32
}
#else
__device__ __forceinline__ float fast_tanh(float x) { return tanhf(x); }
#endif

// ---------------------------------------------------------------------------
// Kernel 1: one block per (b,m). Computes pooled H1,H2 and the squared
// distance d[b,m] = |H1-H2|^2 into workspace.
//
// Score phase per 16-row tile:  Z = Apad(16x16) x X^T(16x16)  via 4 chained
// v_wmma_f32_16x16x4_f32 (A-operand loop-invariant in VGPRs). The D layout
// gives lane (col,half) the components y_l[r+8*half] of y_l = A*x_l for its
// own row l = rbase+col, so S_l = x_l . y_l needs only 8 FMAs + one
// shfl_xor(16) per tile. Then masked softmax over L and weighted pooling.
// ---------------------------------------------------------------------------
__global__ __launch_bounds__(128)
void siamese_attn_pool_kernel(const float* __restrict__ X1,
                              const float* __restrict__ X2,
                              const float* __restrict__ M1,
                              const float* __restrict__ M2,
                              const float* __restrict__ attn_w,
                              float* __restrict__ dist) {
  __shared__ float Xs[NL * XSTR];   // 36 KB staged row (cols 13..15 zeroed)
  __shared__ float Apad[16 * 16];   // attn_w^T zero-padded to 16x16
  __shared__ float Ss[NL];          // scores
  __shared__ float msk[NL];         // mask channel 0
  __shared__ float red[128];        // block reductions
  __shared__ float Hpart[4][16];    // per-wave pooled partials
  __shared__ float Hout[2][16];     // pooled H for tensor 0/1

  const int tid  = threadIdx.x;
  const int lane = tid & 31;
  const int wid  = tid >> 5;
  const int bm   = blockIdx.x;                 // b*NM + m
  const long base = (long)bm * (NL * NFT);     // element offset of (b,m,0,0)

  // Apad[i][j] = A[i][j] = attn_w[j][i], zero padded to 16x16
  for (int idx = tid; idx < 256; idx += 128) {
    int i = idx >> 4, j = idx & 15;
    Apad[idx] = (i < NFT && j < NFT) ? attn_w[j * NFT + i] : 0.0f;
  }
  __syncthreads();

  const int halfsel = lane >> 4;   // 0: lanes 0-15, 1: lanes 16-31
  const int col     = lane & 15;

  // Loop-invariant WMMA A-operands: rows of Apad in the 16x4 A layout.
  // aop[kk] = { Apad[col][kb], Apad[col][kb+1] }, kb = 4*kk + 2*halfsel.
  v2f aop[4];
#pragma unroll
  for (int kk = 0; kk < 4; ++kk) {
    int kb = kk * 4 + halfsel * 2;
    aop[kk] = *reinterpret_cast<const v2f*>(&Apad[col * 16 + kb]);
  }

  for (int t = 0; t < 2; ++t) {
    const float* __restrict__ X  = (t == 0) ? X1 : X2;
    const float* __restrict__ Mk = (t == 0) ? M1 : M2;
    __syncthreads();  // Xs reuse across tensors

    // ---- Stage X row: vectorized global reads, scatter into padded layout.
    // base bytes = bm*26624 -> 16B aligned; NL*NFT divisible by 4.
    const float4* X4 = reinterpret_cast<const float4*>(X + base);
    for (int idx4 = tid; idx4 < (NL * NFT) / 4; idx4 += 128) {
      float4 v = X4[idx4];
      int p = idx4 * 4;
      float vv[4] = {v.x, v.y, v.z, v.w};
#pragma unroll
      for (int e = 0; e < 4; ++e) {
        int pe = p + e;
        int l = pe / NFT;
        int i = pe - l * NFT;
        Xs[l * XSTR + i] = vv[e];
      }
    }
    // Zero K-pad columns; gather mask channel 0 (stride NFT dwords).
    for (int l = tid; l < NL; l += 128) {
      Xs[l * XSTR + 13] = 0.0f;
      Xs[l * XSTR + 14] = 0.0f;
      Xs[l * XSTR + 15] = 0.0f;
      msk[l] = Mk[base + (long)l * NFT];
    }
    __syncthreads();

    // ---- WMMA score phase: uniform trip count (8 tiles per wave).
#pragma unroll 2
    for (int it = 0; it < (NL / 16) / 4; ++it) {
      const int tl    = it * 4 + wid;
      const int row   = tl * 16 + col;      // this lane's sequence position
      const int raddr = row * XSTR;

      v8f acc = {};
#pragma unroll
      for (int kk = 0; kk < 4; ++kk) {
        int kb = kk * 4 + halfsel * 2;      // even -> 8B-aligned ds_load_b64
        v2f b = *reinterpret_cast<const v2f*>(&Xs[raddr + kb]);
        acc = wmma_f32_16x16x4(aop[kk], b, acc);
      }

      // S_row = x_row . y_row : this lane holds y_row[r + 8*halfsel].
      float p = 0.0f;
      const int xoff = raddr + halfsel * 8;
#pragma unroll
      for (int rr = 0; rr < 4; ++rr) {
        v2f xv = *reinterpret_cast<const v2f*>(&Xs[xoff + rr * 2]);
        p += acc[rr * 2] * xv.x;
        p += acc[rr * 2 + 1] * xv.y;
      }
      p += __shfl_xor(p, 16, 32);  // both half-lanes now hold the full dot
      Ss[row] = p;                 // lane pairs write identical values
    }
    __syncthreads();

    // ---- logits = tanh(S) + (mask-1)*1e5 ; softmax over L (4 rows/thread)
    float lg[4];
#pragma unroll
    for (int q = 0; q < 4; ++q) {
      int l = tid + q * 128;
      lg[q] = fast_tanh(Ss[l]) + (msk[l] - 1.0f) * 100000.0f;
    }
    float mx = fmaxf(fmaxf(lg[0], lg[1]), fmaxf(lg[2], lg[3]));
    red[tid] = mx;
    __syncthreads();
    for (int s = 64; s > 0; s >>= 1) {
      if (tid < s) red[tid] = fmaxf(red[tid], red[tid + s]);
      __syncthreads();
    }
    const float gmax = red[0];
    __syncthreads();

    float e[4];
    float psum = 0.0f;
#pragma unroll
    for (int q = 0; q < 4; ++q) {
      e[q] = __expf(lg[q] - gmax);
      psum += e[q];
    }
    red[tid] = psum;
    __syncthreads();
    for (int s = 64; s > 0; s >>= 1) {
      if (tid < s) red[tid] += red[tid + s];
      __syncthreads();
    }
    const float Z = red[0];
    __syncthreads();

    // ---- pooled H_i = (1/Z) * sum_l e_l * X[l][i]
    float acch[NFT];
#pragma unroll
    for (int i = 0; i < NFT; ++i) acch[i] = 0.0f;
#pragma unroll
    for (int q = 0; q < 4; ++q) {
      int l = tid + q * 128;
      float w = e[q];
#pragma unroll
      for (int i = 0; i < NFT; ++i) acch[i] += w * Xs[l * XSTR + i];
    }
#pragma unroll
    for (int i = 0; i < NFT; ++i) {
      float v = acch[i];
      v += __shfl_xor(v, 1, 32);
      v += __shfl_xor(v, 2, 32);
      v += __shfl_xor(v, 4, 32);
      v += __shfl_xor(v, 8, 32);
      v += __shfl_xor(v, 16, 32);
      acch[i] = v;                  // all lanes hold the wave total
    }
    if (lane == 0) {                // one predicated block, 13 stores
#pragma unroll
      for (int i = 0; i < NFT; ++i) Hpart[wid][i] = acch[i];
    }
    __syncthreads();
    if (tid < NFT) {
      Hout[t][tid] =
          (Hpart[0][tid] + Hpart[1][tid] + Hpart[2][tid] + Hpart[3][tid]) / Z;
    }
    __syncthreads();
  }

  if (tid == 0) {
    float d = 0.0f;
#pragma unroll
    for (int i = 0; i < NFT; ++i) {
      float df = Hout[0][i] - Hout[1][i];
      d += df * df;
    }
    dist[bm] = d;
  }
}

// ---------------------------------------------------------------------------
// Kernel 2: out[b][n] = bias[n] + sum_k dist[b][k] * layer_w[n][k]
// Pure WMMA f32 GEMM: M=4 padded into the 16-row A tile, one wave per
// 16-wide N tile, K=1128 in chained 4-wide steps (282 WMMAs/tile).
// ---------------------------------------------------------------------------
__global__ __launch_bounds__(32)
void siamese_head_kernel(const float* __restrict__ dist,
                         const float* __restrict__ layer_w,
                         const float* __restrict__ layer_b,
                         float* __restrict__ out) {
  __shared__ float sdist[NB * NM];  // 18 KB, reused for the whole K loop

  const int lane = threadIdx.x;
  const int n0   = blockIdx.x * 16;

  const float4* d4 = reinterpret_cast<const float4*>(dist);
  for (int idx = lane; idx < (NB * NM) / 4; idx += 32) {
    float4 v = d4[idx];
    sdist[idx * 4 + 0] = v.x;
    sdist[idx * 4 + 1] = v.y;
    sdist[idx * 4 + 2] = v.z;
    sdist[idx * 4 + 3] = v.w;
  }
  __syncthreads();

  const int halfsel = lane >> 4;
  const int col     = lane & 15;
  const int n       = n0 + col;
  const bool nvalid = (n < NM);
  const bool avalid = (col < NB);               // only batch rows 0..3 live
  const int  arow   = avalid ? col : 0;         // clamp for safe b64 loads
  const float* wrow = layer_w + (long)(nvalid ? n : 0) * NM;

  v8f acc = {};
  for (int k = 0; k < NM; k += 4) {             // NM % 4 == 0
    const int koff = k + halfsel * 2;           // even -> 8B aligned
    v2f a = *reinterpret_cast<const v2f*>(&sdist[arow * NM + koff]);
    if (!avalid) { a.x = 0.0f; a.y = 0.0f; }
    v2f b = *reinterpret_cast<const v2f*>(&wrow[koff]);
    if (!nvalid) { b.x = 0.0f; b.y = 0.0f; }
    acc = wmma_f32_16x16x4(a, b, acc);
  }

  // D: vgpr r -> M=r (lanes 0-15) / M=r+8 (lanes 16-31); only M=0..3 valid.
  if (halfsel == 0 && nvalid) {
    const float bias = layer_b[n];
#pragma unroll
    for (int r = 0; r < NB; ++r) {
      out[r * NM + n] = acc[r] + bias;
    }
  }
}

// ---------------------------------------------------------------------------
extern "C" void kernel_launch(void* const* d_in, const int* in_sizes, int n_in,
                              void* d_out, int out_size, void* d_ws,
                              size_t ws_size, hipStream_t stream) {
  (void)in_sizes; (void)n_in; (void)out_size; (void)ws_size;

  const float* X1      = (const float*)d_in[0];
  const float* X2      = (const float*)d_in[1];
  const float* M1      = (const float*)d_in[2];
  const float* M2      = (const float*)d_in[3];
  const float* attn_w  = (const float*)d_in[4];
  const float* layer_w = (const float*)d_in[5];
  const float* layer_b = (const float*)d_in[6];
  float* out  = (float*)d_out;
  float* dist = (float*)d_ws;  // NB*NM floats = 18 KB scratch

  siamese_attn_pool_kernel<<<NB * NM, 128, 0, stream>>>(X1, X2, M1, M2,
                                                        attn_w, dist);
  siamese_head_kernel<<<(NM + 15) / 16, 32, 0, stream>>>(dist, layer_w,
                                                         layer_b, out);
}